// BiDirectionalMultiHeadPointer_22050362097682
// MI455X (gfx1250) — compile-verified
//
#include <hip/hip_runtime.h>
#include <math.h>

// Problem constants (match reference)
#define BB 2
#define NN 1024
#define DD 1024
#define HH 16
#define HD 64
#define BN (BB * NN)          // 2048 rows

// Register blocking for the two big GEMMs (2048x1024x1024)
#define MT 4                  // 16-row tiles per wave  -> 64 rows
#define NT 2                  // 16-col tiles per wave  -> 32 cols

typedef __attribute__((ext_vector_type(2))) float v2f;
typedef __attribute__((ext_vector_type(4))) float v4f;
typedef __attribute__((ext_vector_type(8))) float v8f;

// ---------------------------------------------------------------------------
// One 16-byte A/B fragment -> two V_WMMA_F32_16X16X4_F32 accumulations.
// K-groups are permuted between lane halves (lanes 0-15 carry K {k,k+1} /
// {k+2,k+3}, lanes 16-31 carry {k+4,k+5} / {k+6,k+7}); A and B use the same
// assignment, and the K-contraction is order-free, so the result is exact.
// NOTE: dispatch keys on __has_builtin only -- it is true for the gfx1250
// device pass (verified by emitted v_wmma_f32_16x16x4_f32) and false for the
// host pass, which only needs a parseable stub (never codegen'd).
// ---------------------------------------------------------------------------
__device__ __forceinline__ v8f wmma2_step(v4f a, v4f b, v8f acc) {
#if __has_builtin(__builtin_amdgcn_wmma_f32_16x16x4_f32)
  acc = __builtin_amdgcn_wmma_f32_16x16x4_f32(false, a.xy, false, b.xy,
                                              (short)0, acc, false, false);
  acc = __builtin_amdgcn_wmma_f32_16x16x4_f32(false, a.zw, false, b.zw,
                                              (short)0, acc, false, false);
#else
  acc[0] += a.x * b.x;   // host parse-only stub
#endif
  return acc;
}

// Single-tile K-loop (used by the small per-head K=64 GEMMs).
// 'arow'/'brow' must already include the per-lane +half*4 element offset.
__device__ __forceinline__ v8f wmma_tile_acc(const float* __restrict__ arow,
                                             const float* __restrict__ brow,
                                             int K, v8f acc) {
#pragma unroll 4
  for (int k = 0; k < K; k += 8)
    acc = wmma2_step(*(const v4f*)(arow + k), *(const v4f*)(brow + k), acc);
  return acc;
}

// C/D layout: VGPR i -> lanes 0-15: M=i, lanes 16-31: M=i+8, N = lane&15.
__device__ __forceinline__ void wmma_store(float* __restrict__ c, int ldc, v8f acc) {
  const int l15  = threadIdx.x & 15;
  const int mofs = ((threadIdx.x >> 4) & 1) * 8;
#pragma unroll
  for (int i = 0; i < 8; ++i)
    c[(i + mofs) * ldc + l15] = acc[i];
}

// ---------------------------------------------------------------------------
// Kernel 1/6: C[M,Nout] = A[M,K] * W[Nout,K]^T with 64x32 register blocking.
// Used for HF = h * Wv^T and final = multi * Wo^T.
// Per K-step of 8: 6 x global_load_b128 feed 16 WMMAs (~10.7 FLOP/byte).
// ---------------------------------------------------------------------------
__global__ void k_gemm_nt(const float* __restrict__ A, const float* __restrict__ W,
                          float* __restrict__ C, int K, int Nout) {
  const int wid   = (blockIdx.x * blockDim.x + threadIdx.x) >> 5;
  const int l15   = threadIdx.x & 15;
  const int half4 = ((threadIdx.x >> 4) & 1) * 4;     // K sub-offset of lane half
  const int nBlk  = Nout / (16 * NT);
  const int bm = wid / nBlk;
  const int bn = wid % nBlk;

  const float* arow[MT];
  const float* brow[NT];
#pragma unroll
  for (int i = 0; i < MT; ++i)
    arow[i] = A + (size_t)(bm * 16 * MT + i * 16 + l15) * K + half4;
#pragma unroll
  for (int j = 0; j < NT; ++j)
    brow[j] = W + (size_t)(bn * 16 * NT + j * 16 + l15) * K + half4;

  v8f acc[MT][NT];
#pragma unroll
  for (int i = 0; i < MT; ++i)
#pragma unroll
    for (int j = 0; j < NT; ++j) acc[i][j] = (v8f){};

  for (int k = 0; k < K; k += 8) {
    v4f a[MT], b[NT];
#pragma unroll
    for (int i = 0; i < MT; ++i) a[i] = *(const v4f*)(arow[i] + k);
#pragma unroll
    for (int j = 0; j < NT; ++j) b[j] = *(const v4f*)(brow[j] + k);
#pragma unroll
    for (int i = 0; i < MT; ++i)
#pragma unroll
      for (int j = 0; j < NT; ++j)
        acc[i][j] = wmma2_step(a[i], b[j], acc[i][j]);
  }

#pragma unroll
  for (int i = 0; i < MT; ++i)
#pragma unroll
    for (int j = 0; j < NT; ++j)
      wmma_store(C + (size_t)(bm * 16 * MT + i * 16) * Nout + bn * 16 * NT + j * 16,
                 Nout, acc[i][j]);
}

// ---------------------------------------------------------------------------
// Kernel 2: MEAN[b, c] = (1/N) sum_n HF[b*N+n, c]   (== fwd_feats == bwd_feats,
// since softmax of a broadcast scalar is exactly uniform 1/N in fp32)
// ---------------------------------------------------------------------------
__global__ void k_colmean(const float* __restrict__ HF, float* __restrict__ MEAN) {
  const int t = blockIdx.x * blockDim.x + threadIdx.x;
  if (t >= BB * DD) return;
  const int b = t / DD, c = t % DD;
  const float* p = HF + (size_t)b * NN * DD + c;
  float s = 0.f;
  for (int n = 0; n < NN; ++n) s += p[(size_t)n * DD];
  MEAN[t] = s * (1.0f / NN);
}

// ---------------------------------------------------------------------------
// Kernel 3: fold the (constant-over-n) fwd/bwd feature contribution into a
// per-(b, head, e') bias: B2 = mean_h * (W1[:,64:128] + W1[:,128:192])^T + b1
// ---------------------------------------------------------------------------
__global__ void k_bias2(const float* __restrict__ MEAN, const float* __restrict__ W1,
                        const float* __restrict__ b1, float* __restrict__ B2) {
  const int t = blockIdx.x * blockDim.x + threadIdx.x;
  if (t >= BB * DD) return;
  const int b = t / DD, g = t % DD, head = g / HD, ep = g % HD;
  const float* m = MEAN + b * DD + head * HD;
  const float* w = W1 + (size_t)(head * HD + ep) * (3 * HD);
  float s = b1[g];
  for (int c = 0; c < HD; ++c) s += m[c] * (w[HD + c] + w[2 * HD + c]);
  B2[t] = s;
}

__device__ __forceinline__ float gelu_exact(float x) {
  return 0.5f * x * (1.0f + erff(x * 0.70710678118654752f));
}

// ---------------------------------------------------------------------------
// Kernel 4: per-head fusion stage 1 (K=64 WMMA) + exact-erf GELU
// X[r, h*64+e'] = gelu( HF[r, h*64 : +64] . fu_W1[h, e', 0:64] + B2[b, h*64+e'] )
// ---------------------------------------------------------------------------
__global__ void k_fuse1(const float* __restrict__ HF, const float* __restrict__ W1,
                        const float* __restrict__ B2, float* __restrict__ X) {
  const int wid   = (blockIdx.x * blockDim.x + threadIdx.x) >> 5;
  const int l15   = threadIdx.x & 15;
  const int half4 = ((threadIdx.x >> 4) & 1) * 4;
  const int tilesN = DD >> 4;                 // 64 column tiles, 4 per head
  const int tm = wid / tilesN;
  const int tn = wid % tilesN;
  const int head = tn >> 2;
  const int gcol = tn * 16 + l15;             // global output column
  const int ep   = gcol & (HD - 1);           // e' inside head
  const float* arow = HF + (size_t)(tm * 16 + l15) * DD + head * HD + half4;
  const float* brow = W1 + (size_t)(head * HD + ep) * (3 * HD) + half4; // c in [0,64)
  v8f acc = {};
  acc = wmma_tile_acc(arow, brow, HD, acc);
  const int b = (tm * 16) / NN;               // whole tile shares one batch index
  const float bias = B2[b * DD + gcol];
  float* c = X + (size_t)tm * 16 * DD + tn * 16;
  const int mofs = ((threadIdx.x >> 4) & 1) * 8;
#pragma unroll
  for (int i = 0; i < 8; ++i)
    c[(i + mofs) * DD + l15] = gelu_exact(acc[i] + bias);
}

// ---------------------------------------------------------------------------
// Kernel 5: per-head fusion stage 2 (K=64 WMMA):
// MULT[r, h*64+f] = X[r, h*64 : +64] . fu_W2[h, f, :] + fu_b2[h, f]
// (MULT already has the (b, n, h, f) -> (b, n, d) interleave of the reference)
// ---------------------------------------------------------------------------
__global__ void k_fuse2(const float* __restrict__ X, const float* __restrict__ W2,
                        const float* __restrict__ b2, float* __restrict__ MULT) {
  const int wid   = (blockIdx.x * blockDim.x + threadIdx.x) >> 5;
  const int l15   = threadIdx.x & 15;
  const int half4 = ((threadIdx.x >> 4) & 1) * 4;
  const int tilesN = DD >> 4;
  const int tm = wid / tilesN;
  const int tn = wid % tilesN;
  const int head = tn >> 2;
  const int gcol = tn * 16 + l15;
  const int f    = gcol & (HD - 1);
  const float* arow = X + (size_t)(tm * 16 + l15) * DD + head * HD + half4;
  const float* brow = W2 + (size_t)(head * HD + f) * HD + half4;
  v8f acc = {};
  acc = wmma_tile_acc(arow, brow, HD, acc);
  const float bias = b2[gcol];                // fu_b2 flat [H*HD] == column index
  float* c = MULT + (size_t)tm * 16 * DD + tn * 16;
  const int mofs = ((threadIdx.x >> 4) & 1) * 8;
#pragma unroll
  for (int i = 0; i < 8; ++i)
    c[(i + mofs) * DD + l15] = acc[i] + bias;
}

// ---------------------------------------------------------------------------
// Kernel 6: targets + strength (exact constants from the uniform softmax).
// ---------------------------------------------------------------------------
__global__ void k_tail(const int* __restrict__ prev_idx, const float* __restrict__ cr,
                       float* __restrict__ out) {
  const int t = blockIdx.x * blockDim.x + threadIdx.x;   // t = b*N + n
  if (t >= BB * NN) return;
  const int n = t % NN;
  const float sig = 1.0f / (1.0f + expf(-cr[0]));
  const float thr = floorf(sig * (float)NN);             // 637 for cr = 0.5
  const float uni = (float)((NN - 1) / 2);               // sum(m/N) = 511.5 -> int 511
  int pc = prev_idx[t];
  pc = pc < 0 ? 0 : (pc > NN - 1 ? NN - 1 : pc);
  const float fwd = ((float)n >= thr) ? (float)pc : uni;
  const float strength = 1.0f - logf(1.0f / (float)NN + 1e-8f);
  float* base = out + (size_t)BN * DD;                   // after final_output
  base[t]               = fwd;        // fwd_targets[0]  [B,N]
  base[BB * NN + t]     = uni;        // bwd_targets[0]  [B,N]
  base[2 * BB * NN + t] = strength;   // avg_strength    [B,N]
}

// ---------------------------------------------------------------------------
extern "C" void kernel_launch(void* const* d_in, const int* in_sizes, int n_in,
                              void* d_out, int out_size, void* d_ws, size_t ws_size,
                              hipStream_t stream) {
  (void)in_sizes; (void)n_in; (void)out_size; (void)ws_size;
  const float* h    = (const float*)d_in[0];
  const int*   prev = (const int*)d_in[1];
  // d_in[2..9] = fw_*/bw_* encoder weights: provably dead (uniform softmax).
  const float* Wv   = (const float*)d_in[10];   // [H,HD,D] == [1024, 1024] row-major
  const float* fuW1 = (const float*)d_in[11];   // [H,HD,3HD]
  const float* fub1 = (const float*)d_in[12];   // [H,HD]
  const float* fuW2 = (const float*)d_in[13];   // [H,HD,HD]
  const float* fub2 = (const float*)d_in[14];   // [H,HD]
  const float* Wo   = (const float*)d_in[15];   // [D,D]
  const float* cr   = (const float*)d_in[16];   // scalar

  float* out  = (float*)d_out;
  float* HF   = (float*)d_ws;           // [BN, D]  head_features
  float* X    = HF + (size_t)BN * DD;   // [BN, D]  gelu(fusion stage 1)
  float* MULT = X  + (size_t)BN * DD;   // [BN, D]  head_out interleaved
  float* MEAN = MULT + (size_t)BN * DD; // [B, D]
  float* B2   = MEAN + (size_t)BB * DD; // [B, D]

  const dim3 blk(256);                                     // 8 wave32 per block
  const int gemm_waves = (BN / (16 * MT)) * (DD / (16 * NT));   // 32*32 = 1024
  const dim3 grid_gemm(gemm_waves / 8);                    // 128 blocks
  const int fuse_waves = (BN / 16) * (DD / 16);            // 8192 16x16 tiles
  const dim3 grid_fuse(fuse_waves / 8);                    // 1024 blocks
  const dim3 grid_ew((BB * DD + 255) / 256);               // 2048 threads

  k_gemm_nt<<<grid_gemm, blk, 0, stream>>>(h, Wv, HF, DD, DD);       // HF = h Wv^T
  k_colmean<<<grid_ew, blk, 0, stream>>>(HF, MEAN);                  // mean over n
  k_bias2  <<<grid_ew, blk, 0, stream>>>(MEAN, fuW1, fub1, B2);      // fold fwd/bwd feats
  k_fuse1  <<<grid_fuse, blk, 0, stream>>>(HF, fuW1, B2, X);         // K=64 WMMA + GELU
  k_fuse2  <<<grid_fuse, blk, 0, stream>>>(X, fuW2, fub2, MULT);     // K=64 WMMA
  k_gemm_nt<<<grid_gemm, blk, 0, stream>>>(MULT, Wo, out, DD, DD);   // final = multi Wo^T
  k_tail   <<<grid_ew, blk, 0, stream>>>(prev, cr, out);             // targets + strength
}